// SmoothLoss_62594853372240
// MI455X (gfx1250) — compile-verified
//
#include <hip/hip_runtime.h>

// SmoothLoss: sum over rows of (A[i]-A[i+1])^2, A is [1024, 32768] fp32.
// Memory-bound: 128 MiB read, ~100 MFLOP -> HBM floor ~5.8us @ 23.3 TB/s.
// Input fits in the 192MB global L2, so timed replays may run at L2 speed:
// keep default RT temporal hints (no NT/LU that would bypass/evict L2).

typedef float v2f __attribute__((ext_vector_type(2)));
typedef float v8f __attribute__((ext_vector_type(8)));

#define ROW_LEN    32768u          // T from the reference shape (1024 x 32768)
#define ROW_VECS   (ROW_LEN / 4u)  // 8192 float4 vectors per row (power of 2)
#define N_BLOCKS   1024            // kernel-1 blocks == number of partials
#define N_THREADS  256             // 8 waves/block under wave32
#define FIN_THREADS (N_BLOCKS / 4) // kernel-2: one float4 of partials per thread

// Per-vector work: 3 intra-vector pairs + 1 cross-vector pair via wave32
// shuffle (lanes hold consecutive vectors); lane 31 patches with a 4B load.
__device__ __forceinline__ float process_vec(const float4* __restrict__ A4,
                                             const float*  __restrict__ A,
                                             unsigned int v, unsigned int lane,
                                             unsigned int N) {
    float4 x = A4[v];
    float d0 = x.x - x.y;
    float d1 = x.y - x.z;
    float d2 = x.z - x.w;
    float acc = d0 * d0 + d1 * d1 + d2 * d2;

    float nx = __shfl_down(x.x, 1, 32);
    if (lane == 31u) {                        // wave boundary: one 4B patch load
        unsigned int e = 4u * v + 4u;
        nx = A[e < N ? e : (N - 1u)];
    }
    // Skip the cross-vector pair at the end of each row (vectors never
    // straddle rows since ROW_LEN % 4 == 0). ROW_VECS is a power of two.
    if ((v & (ROW_VECS - 1u)) != (ROW_VECS - 1u)) {
        float d3 = x.w - nx;
        acc += d3 * d3;
    }
    return acc;
}

// Kernel 1: grid-stride b128 streaming, unrolled x2 so two independent
// global_load_b128 are in flight per thread before any wait. One fp32
// partial per block into d_ws, fixed order -> deterministic.
__global__ __launch_bounds__(N_THREADS)
void smooth_partial_kernel(const float* __restrict__ A,
                           float* __restrict__ partial,
                           unsigned int V,      // number of float4 vectors
                           unsigned int N) {    // number of floats
    const unsigned int lane   = threadIdx.x & 31u;
    const unsigned int tid    = blockIdx.x * N_THREADS + threadIdx.x;
    const unsigned int stride = gridDim.x * N_THREADS;
    const float4* __restrict__ A4 = (const float4*)A;

    float acc0 = 0.0f, acc1 = 0.0f;
    unsigned int v = tid;
    for (; v + stride < V; v += 2u * stride) {
        // Speculative prefetch of a future chunk (global_prefetch_b8).
        __builtin_prefetch(&A4[v + 2u * stride], 0, 3);
        acc0 += process_vec(A4, A, v,          lane, N);
        acc1 += process_vec(A4, A, v + stride, lane, N);
    }
    if (v < V) acc0 += process_vec(A4, A, v, lane, N);
    float acc = acc0 + acc1;

    // In-wave tree reduction (wave32).
    for (int off = 16; off > 0; off >>= 1)
        acc += __shfl_down(acc, off, 32);

    __shared__ float wsum[N_THREADS / 32];
    if (lane == 0u) wsum[threadIdx.x >> 5] = acc;
    __syncthreads();
    if (threadIdx.x == 0u) {
        float s = 0.0f;
        for (int i = 0; i < N_THREADS / 32; ++i) s += wsum[i];
        partial[blockIdx.x] = s;
    }
}

// Kernel 2: parallel finalize, exactly ONE b128 load round-trip per thread
// (1024 partials = 256 float4 across 256 threads), deterministic fixed-order
// reduction, no atomics. Wave 0 folds the 8 wave-sums with
// V_WMMA_F32_16X16X4_F32: with B = all-ones, sum of all 256 D entries equals
// 16 * sum(A entries) regardless of the lane<->slot mapping, and /16 is exact.
__global__ __launch_bounds__(FIN_THREADS)
void smooth_final_kernel(const float* __restrict__ partial,
                         float* __restrict__ out) {
    const unsigned int tid  = threadIdx.x;
    const unsigned int lane = tid & 31u;
    const float4* __restrict__ P4 = (const float4*)partial;

    float4 p = P4[tid];                       // single memory round-trip
    float s = (p.x + p.y) + (p.z + p.w);

    for (int off = 16; off > 0; off >>= 1)
        s += __shfl_down(s, off, 32);

    __shared__ float wsum[FIN_THREADS / 32];
    if (lane == 0u) wsum[tid >> 5] = s;
    __syncthreads();

    if (tid < 32u) {                          // wave 0, EXEC all ones
        float vv = (lane < (FIN_THREADS / 32u)) ? wsum[lane] : 0.0f;

        v2f a; a[0] = vv;   a[1] = 0.0f;      // wave sums occupy <=32 A slots
        v2f b; b[0] = 1.0f; b[1] = 1.0f;      // all 64 B slots = 1.0
        v8f c = {};
        v8f d = __builtin_amdgcn_wmma_f32_16x16x4_f32(
            /*neg_a=*/false, a, /*neg_b=*/false, b,
            /*c_mod=*/(short)0, c, /*reuse_a=*/false, /*reuse_b=*/false);

        float t = ((d[0] + d[1]) + (d[2] + d[3])) +
                  ((d[4] + d[5]) + (d[6] + d[7]));
        for (int off = 16; off > 0; off >>= 1)
            t += __shfl_down(t, off, 32);

        if (lane == 0u) out[0] = t * 0.0625f; // exact /16 (power of two)
    }
}

extern "C" void kernel_launch(void* const* d_in, const int* in_sizes, int n_in,
                              void* d_out, int out_size, void* d_ws, size_t ws_size,
                              hipStream_t stream) {
    const float* A = (const float*)d_in[0];
    const unsigned int N = (unsigned int)in_sizes[0];  // 1024*32768
    const unsigned int V = N / 4u;

    float* partial = (float*)d_ws;   // N_BLOCKS * 4 bytes = 4 KB scratch
    float* out     = (float*)d_out;

    smooth_partial_kernel<<<N_BLOCKS, N_THREADS, 0, stream>>>(A, partial, V, N);
    smooth_final_kernel<<<1, FIN_THREADS, 0, stream>>>(partial, out);
}